// Lag_51866025066809
// MI455X (gfx1250) — compile-verified
//
#include <hip/hip_runtime.h>

// Problem geometry: z[N=8192, B=64, D=128] f32 -> cumsum along N.
// Flattened as M[n, c], c = b*128 + d, row stride NC = 8192 floats.
#define NROWS  8192
#define NC     8192            // 64*128 columns
#define NCOL4  (NC / 4)        // 2048 float4 columns
#define CROWS  64              // rows per chunk
#define NCHUNK (NROWS / CROWS) // 128 chunks
#define NTILE  (NCHUNK / 16)   // 8 16-chunk tiles in pass 2

typedef __attribute__((ext_vector_type(2))) float v2f;
typedef __attribute__((ext_vector_type(4))) float v4f;
typedef __attribute__((ext_vector_type(8))) float v8f;

// ---------------- Pass 1: per-(chunk, column) sums -> ws ----------------
// 262144 threads; thread t: col4 = t % 2048, chunk = t / 2048.
// Wave loads 32 * 16B = 512B contiguous per row step: fully coalesced.
__global__ __launch_bounds__(256) void lag_pass1(const float* __restrict__ z,
                                                 float* __restrict__ sums) {
  int t = blockIdx.x * 256 + threadIdx.x;
  int col4 = t & (NCOL4 - 1);
  int chunk = t >> 11;
  const v4f* pz = (const v4f*)z + (size_t)chunk * CROWS * NCOL4 + col4;
  v4f acc = {0.0f, 0.0f, 0.0f, 0.0f};
#pragma unroll 8
  for (int r = 0; r < CROWS; ++r) acc += pz[(size_t)r * NCOL4];
  ((v4f*)sums)[(size_t)chunk * NCOL4 + col4] = acc;
}

// ---------------- Pass 2: inclusive scan of chunk sums (WMMA) ----------------
// Inclusive prefix sum over the 128-chunk axis, per column, done as
// D = L * S (+ carry) with L = 16x16 lower-triangular ones, using 4 chained
// v_wmma_f32_16x16x4_f32 per 16x16 tile (K=16 split into 4 K-slices).
// Layouts per CDNA5 ISA 7.12.2 (f32):
//   A 16x4 : lane l holds row m=l%16; VGPR0 = K=(l<16?0:2), VGPR1 = K+1
//   B 4x16 : lane l holds col n=l%16; VGPR0 = K=(l<16?0:2), VGPR1 = K+1
//   C/D    : lane l holds col l%16; VGPR v = row v + 8*(l>=16)
// Carry across tiles = D row 15 = VGPR7 of lane 16+(col%16), via __shfl.
// One wave per 16-column group (512 waves total). In-place on `sums`.
__global__ __launch_bounds__(256) void lag_pass2(float* __restrict__ sums) {
  int lane = threadIdx.x & 31;
  int wave = (blockIdx.x * 256 + threadIdx.x) >> 5;  // 0..511
  int m = lane & 15;
  int kh = (lane & 16) ? 2 : 0;
  int col = (wave << 4) + m;

#if __has_builtin(__builtin_amdgcn_wmma_f32_16x16x4_f32)
  // A-matrix K-slices of lower-triangular ones (inclusive: k <= m).
  v2f A0, A1, A2, A3;
  A0.x = (0 + kh <= m) ? 1.0f : 0.0f;   A0.y = (1 + kh <= m) ? 1.0f : 0.0f;
  A1.x = (4 + kh <= m) ? 1.0f : 0.0f;   A1.y = (5 + kh <= m) ? 1.0f : 0.0f;
  A2.x = (8 + kh <= m) ? 1.0f : 0.0f;   A2.y = (9 + kh <= m) ? 1.0f : 0.0f;
  A3.x = (12 + kh <= m) ? 1.0f : 0.0f;  A3.y = (13 + kh <= m) ? 1.0f : 0.0f;

  float carry = 0.0f;
  int hi = (lane >> 4) & 1;
#pragma unroll 1
  for (int tt = 0; tt < NTILE; ++tt) {
    int rb = tt * 16;
    v8f c;
#pragma unroll
    for (int i = 0; i < 8; ++i) c[i] = carry;  // inject carry via accumulator

    v2f b0, b1, b2, b3;  // B layout loads (K-slices of S tile rows)
    b0.x = sums[(size_t)(rb + 0 + kh) * NC + col];
    b0.y = sums[(size_t)(rb + 1 + kh) * NC + col];
    b1.x = sums[(size_t)(rb + 4 + kh) * NC + col];
    b1.y = sums[(size_t)(rb + 5 + kh) * NC + col];
    b2.x = sums[(size_t)(rb + 8 + kh) * NC + col];
    b2.y = sums[(size_t)(rb + 9 + kh) * NC + col];
    b3.x = sums[(size_t)(rb + 12 + kh) * NC + col];
    b3.y = sums[(size_t)(rb + 13 + kh) * NC + col];

    c = __builtin_amdgcn_wmma_f32_16x16x4_f32(false, A0, false, b0, (short)0, c, false, false);
    c = __builtin_amdgcn_wmma_f32_16x16x4_f32(false, A1, false, b1, (short)0, c, false, false);
    c = __builtin_amdgcn_wmma_f32_16x16x4_f32(false, A2, false, b2, (short)0, c, false, false);
    c = __builtin_amdgcn_wmma_f32_16x16x4_f32(false, A3, false, b3, (short)0, c, false, false);

#pragma unroll
    for (int v = 0; v < 8; ++v)
      sums[(size_t)(rb + v + 8 * hi) * NC + col] = c[v];  // C/D layout store

    // new carry = global inclusive value at tile row 15 (already includes old carry)
    carry = __shfl(c[7], 16 + m, 32);
  }
#else
  // Scalar fallback (compile-safety): lanes 0..15 scan one column each.
  if ((lane & 16) == 0) {
    float acc = 0.0f;
    for (int k = 0; k < NCHUNK; ++k) {
      size_t idx = (size_t)k * NC + col;
      acc += sums[idx];
      sums[idx] = acc;
    }
  }
#endif
}

// ---------------- Pass 3: local scan + carry, streamed out ----------------
__global__ __launch_bounds__(256) void lag_pass3(const float* __restrict__ z,
                                                 const float* __restrict__ scan,
                                                 float* __restrict__ out) {
  int t = blockIdx.x * 256 + threadIdx.x;
  int col4 = t & (NCOL4 - 1);
  int chunk = t >> 11;
  size_t base = (size_t)chunk * CROWS * NCOL4 + col4;
  const v4f* pz = (const v4f*)z + base;
  v4f* po = (v4f*)out + base;
  v4f acc = {0.0f, 0.0f, 0.0f, 0.0f};
  if (chunk > 0) acc = ((const v4f*)scan)[(size_t)(chunk - 1) * NCOL4 + col4];
#pragma unroll 8
  for (int r = 0; r < CROWS; ++r) {
    acc += pz[(size_t)r * NCOL4];
    // Output is write-once: non-temporal store keeps L2 for the input stream.
    __builtin_nontemporal_store(acc, po + (size_t)r * NCOL4);
  }
}

// ---------------- previous_loss passthrough ----------------
__global__ void lag_copy_loss(const float* __restrict__ pl, float* __restrict__ out,
                              unsigned long long off) {
  out[off] = pl[0];
}

extern "C" void kernel_launch(void* const* d_in, const int* in_sizes, int n_in,
                              void* d_out, int out_size, void* d_ws, size_t ws_size,
                              hipStream_t stream) {
  const float* z = (const float*)d_in[0];         // [8192, 64, 128] f32
  const float* pl = (const float*)d_in[1];        // [1] f32
  float* out = (float*)d_out;                     // [8192*64*128 + 1] f32
  float* sums = (float*)d_ws;                     // NCHUNK*NC floats = 4 MiB scratch

  const int threads13 = NCHUNK * NCOL4;           // 262144
  lag_pass1<<<threads13 / 256, 256, 0, stream>>>(z, sums);
  lag_pass2<<<(512 * 32) / 256, 256, 0, stream>>>(sums);   // 512 waves
  lag_pass3<<<threads13 / 256, 256, 0, stream>>>(z, sums, out);
  lag_copy_loss<<<1, 1, 0, stream>>>(pl, out, (unsigned long long)in_sizes[0]);
}